// RecallWindowObserver_31052613550198
// MI455X (gfx1250) — compile-verified
//
#include <hip/hip_runtime.h>
#include <cfloat>

#define ROW_N   8192
#define TARGET  7680                    // int(0.9375 * 8192)
#define NCAND   (ROW_N - TARGET + 1)    // 513 candidate windows
#define THREADS 512                     // 16 waves of 32
#define ELEMS   (ROW_N / THREADS)       // 16 contiguous elements per thread

// CDNA5 async global -> LDS copy (GVS mode: saddr base + 32-bit vaddr offset).
// Tracked by ASYNCcnt; completion enforced with s_wait_asynccnt.
__device__ __forceinline__ void async_load_b128(unsigned lds_off, unsigned g_off,
                                                const void* base) {
  asm volatile("global_load_async_to_lds_b128 %0, %1, %2 offset:0"
               :
               : "v"(lds_off), "v"(g_off), "s"(base)
               : "memory");
}

__device__ __forceinline__ void wait_asynccnt0() {
  asm volatile("s_wait_asynccnt 0x0" ::: "memory");
}

// Compare-exchange: after the call, (a,b) sorted ascending if up, else descending.
__device__ __forceinline__ void ce(float& a, float& b, bool up) {
  float lo = fminf(a, b);
  float hi = fmaxf(a, b);
  a = up ? lo : hi;
  b = up ? hi : lo;
}

__global__ __launch_bounds__(THREADS)
void RecallWindow_kernel(const float* __restrict__ x, float* __restrict__ out,
                         int rows) {
  __shared__ float s[ROW_N];        // 32 KB: one full row, sorted in place
  __shared__ float rlen[THREADS];   // argmin reduction: window lengths
  __shared__ int   ridx[THREADS];   // argmin reduction: window indices

  const int row = blockIdx.x;       // uniform per block
  if (row >= rows) return;
  const unsigned t = threadIdx.x;
  const float* rowp = x + (size_t)row * ROW_N;

  // ---- Stream the row into LDS via the async copy engine ----------------
  unsigned ldsBase = (unsigned)(uintptr_t)(&s[0]);
#pragma unroll
  for (int c = 0; c < 4; ++c) {
    unsigned off = (unsigned)(c * THREADS + t) * 16u;   // byte offset, coalesced
    async_load_b128(ldsBase + off, off, (const void*)rowp);
  }
  wait_asynccnt0();
  __syncthreads();

  const unsigned base = t * ELEMS;          // this thread's contiguous block
  float4* const sv = (float4*)&s[base];

  // ---- Phase A: k = 2,4,8,16 entirely in registers ----------------------
  {
    float r[ELEMS];
    float4 v0 = sv[0], v1 = sv[1], v2 = sv[2], v3 = sv[3];
    r[0]=v0.x; r[1]=v0.y; r[2]=v0.z; r[3]=v0.w;
    r[4]=v1.x; r[5]=v1.y; r[6]=v1.z; r[7]=v1.w;
    r[8]=v2.x; r[9]=v2.y; r[10]=v2.z; r[11]=v2.w;
    r[12]=v3.x; r[13]=v3.y; r[14]=v3.z; r[15]=v3.w;
#pragma unroll
    for (unsigned k = 2; k <= 16; k <<= 1) {
#pragma unroll
      for (unsigned j = k >> 1; j > 0; j >>= 1) {
#pragma unroll
        for (unsigned e = 0; e < ELEMS; ++e) {
          if ((e & j) == 0) ce(r[e], r[e + j], (((base + e) & k) == 0));
        }
      }
    }
    v0 = make_float4(r[0], r[1], r[2], r[3]);
    v1 = make_float4(r[4], r[5], r[6], r[7]);
    v2 = make_float4(r[8], r[9], r[10], r[11]);
    v3 = make_float4(r[12], r[13], r[14], r[15]);
    sv[0]=v0; sv[1]=v1; sv[2]=v2; sv[3]=v3;
  }
  __syncthreads();

  // ---- Phase B: k = 32..8192 -------------------------------------------
  for (unsigned k = 32; k <= ROW_N; k <<= 1) {
    // Cross passes (j >= 16): direction & partner constant per quad -> b128 CEs.
    for (unsigned j = k >> 1; j >= 16; j >>= 1) {
      const unsigned jq = j >> 2;            // partner distance in quads
#pragma unroll
      for (unsigned m = 0; m < 2; ++m) {
        unsigned idxq = t + m * THREADS;                       // 0..1023
        unsigned q  = ((idxq & ~(jq - 1)) << 1) | (idxq & (jq - 1)); // (q & jq)==0
        unsigned qp = q + jq;
        bool up = (((q * 4) & k) == 0);
        float4 a = *(float4*)&s[q * 4];
        float4 b = *(float4*)&s[qp * 4];
        ce(a.x, b.x, up); ce(a.y, b.y, up);
        ce(a.z, b.z, up); ce(a.w, b.w, up);
        *(float4*)&s[q * 4]  = a;
        *(float4*)&s[qp * 4] = b;
      }
      __syncthreads();
    }
    // Local tail (j = 8,4,2,1): in registers; direction uniform per thread.
    {
      float r[ELEMS];
      float4 v0 = sv[0], v1 = sv[1], v2 = sv[2], v3 = sv[3];
      r[0]=v0.x; r[1]=v0.y; r[2]=v0.z; r[3]=v0.w;
      r[4]=v1.x; r[5]=v1.y; r[6]=v1.z; r[7]=v1.w;
      r[8]=v2.x; r[9]=v2.y; r[10]=v2.z; r[11]=v2.w;
      r[12]=v3.x; r[13]=v3.y; r[14]=v3.z; r[15]=v3.w;
      const bool up = ((base & k) == 0);   // k >= 32: constant over 16 elems
#pragma unroll
      for (unsigned j = 8; j > 0; j >>= 1) {
#pragma unroll
        for (unsigned e = 0; e < ELEMS; ++e) {
          if ((e & j) == 0) ce(r[e], r[e + j], up);
        }
      }
      v0 = make_float4(r[0], r[1], r[2], r[3]);
      v1 = make_float4(r[4], r[5], r[6], r[7]);
      v2 = make_float4(r[8], r[9], r[10], r[11]);
      v3 = make_float4(r[12], r[13], r[14], r[15]);
      sv[0]=v0; sv[1]=v1; sv[2]=v2; sv[3]=v3;
    }
    __syncthreads();
  }

  // ---- Minimum-length window over 513 candidates, first-min semantics ---
  float bestLen = FLT_MAX;
  int   bestIdx = 0;
  for (int i = (int)t; i < NCAND; i += THREADS) {   // t==0 also handles i=512
    float len = s[i + TARGET - 1] - s[i];
    if (len < bestLen) { bestLen = len; bestIdx = i; }  // strict '<': keep earliest
  }
  rlen[t] = bestLen;
  ridx[t] = bestIdx;
  __syncthreads();

  for (int off = THREADS / 2; off > 0; off >>= 1) {
    if ((int)t < off) {
      float l2 = rlen[t + off];
      int   i2 = ridx[t + off];
      float l1 = rlen[t];
      int   i1 = ridx[t];
      if (l2 < l1 || (l2 == l1 && i2 < i1)) { rlen[t] = l2; ridx[t] = i2; }
    }
    __syncthreads();
  }

  if (t == 0) {
    int bi = ridx[0];
    out[row]        = s[bi];                 // left
    out[rows + row] = s[bi + TARGET - 1];    // right
  }
}

extern "C" void kernel_launch(void* const* d_in, const int* in_sizes, int n_in,
                              void* d_out, int out_size, void* d_ws, size_t ws_size,
                              hipStream_t stream) {
  const float* x = (const float*)d_in[0];
  float* out = (float*)d_out;
  const int rows = in_sizes[0] / ROW_N;   // 4096 for the reference shapes
  RecallWindow_kernel<<<rows, THREADS, 0, stream>>>(x, out, rows);
}